// MultilabelKnnModule_54975581389396
// MI455X (gfx1250) — compile-verified
//
#include <hip/hip_runtime.h>
#include <hip/hip_bf16.h>

typedef float v2f __attribute__((ext_vector_type(2)));
typedef float v8f __attribute__((ext_vector_type(8)));

#define NCLS    19
#define K_A     10
#define K_B     20
#define MAXK    20
#define DFEAT   128
#define NTRAIN  16384
#define NQUERY  4096
#define FLT_BIG 3.4e38f

// -------------------------------------------------------------------------
// Row squared-norms: one thread per row, float4 loads.
// -------------------------------------------------------------------------
__global__ void row_norms_kernel(const float* __restrict__ X,
                                 float* __restrict__ out, int nrows) {
    int r = blockIdx.x * blockDim.x + threadIdx.x;
    if (r >= nrows) return;
    const float4* p = (const float4*)(X + (size_t)r * DFEAT);
    float s = 0.f;
#pragma unroll
    for (int i = 0; i < DFEAT / 4; ++i) {
        float4 v = p[i];
        s += v.x * v.x + v.y * v.y + v.z * v.z + v.w * v.w;
    }
    out[r] = s;
}

// -------------------------------------------------------------------------
// Transpose train features into k-PAIR-INTERLEAVED layout:
//   TTI[p][n][i] = T[n][2p + i],  p = 0..DFEAT/2-1, i = 0..1
// so a WMMA B fragment (K=k0,k0+1 at column n) is ONE aligned b64 load.
// -------------------------------------------------------------------------
__global__ void interleave_kernel(const float* __restrict__ T,
                                  float* __restrict__ TTI) {
    int idx = blockIdx.x * blockDim.x + threadIdx.x;  // idx = p*NTRAIN + n
    if (idx >= (DFEAT / 2) * NTRAIN) return;
    int p = idx / NTRAIN;
    int n = idx - p * NTRAIN;
    float2 v = *(const float2*)&T[(size_t)n * DFEAT + 2 * p];
    *(float2*)&TTI[((size_t)p * NTRAIN + n) * 2] = v;
}

// -------------------------------------------------------------------------
// Fused distance GEMM (V_WMMA_F32_16X16X4_F32) + streaming top-20 per row.
// Block: 256 threads = 8 wave32; owns 16 query rows.
// A tile staged in LDS via GLOBAL_LOAD_ASYNC_TO_LDS_B128 (ASYNCcnt).
// Inner loop: 1 ds_load_b64 (A pair) + 1 global_load_b64 (B pair) + 1 wmma.
// Two-level top-k: 16 threads/row keep private sorted top-20 over an
// 8-column segment; merged once at the end.
// -------------------------------------------------------------------------
__launch_bounds__(256)
__global__ void knn_topk_kernel(const float* __restrict__ Q,
                                const float* __restrict__ qnorm,
                                const float* __restrict__ TTI,   // [DFEAT/2][NTRAIN][2]
                                const float* __restrict__ tnorm, // [NTRAIN]
                                int* __restrict__ nbr) {         // [rows][MAXK]
    __shared__ float As[16][DFEAT + 4];   // stride 132 floats = 528 B (16B-aligned)
    __shared__ float qn[16];
    __shared__ float Dist[16][128];
    __shared__ float pld[256][MAXK];
    __shared__ int   pli[256][MAXK];
    __shared__ float topd[16][MAXK];
    __shared__ int   topi[16][MAXK];

    const int tid  = threadIdx.x;
    const int wave = tid >> 5;
    const int lane = tid & 31;
    const int half = lane >> 4;     // 0: lanes 0-15, 1: lanes 16-31
    const int n16  = lane & 15;
    const int rowblk = blockIdx.x * 16;

    // ---- async copy A tile (16 x 128 f32) into LDS: 512 x b128 transfers ----
    for (int i = tid; i < 16 * (DFEAT / 4); i += 256) {
        int m  = i >> 5;            // row 0..15
        int kq = (i & 31) * 4;      // dword-quad offset, 16B aligned
        unsigned lds  = (unsigned)(size_t)&As[m][kq];
        unsigned goff = (unsigned)(((rowblk + m) * DFEAT + kq) * sizeof(float));
        asm volatile("global_load_async_to_lds_b128 %0, %1, %2"
                     :: "v"(lds), "v"(goff), "s"(Q) : "memory");
    }
    asm volatile("s_wait_asynccnt 0x0" ::: "memory");

    if (tid < 16) qn[tid] = qnorm[rowblk + tid];
#pragma unroll
    for (int j = 0; j < MAXK; ++j) { pld[tid][j] = FLT_BIG; pli[tid][j] = 0; }
    __syncthreads();

    const int selrow = tid & 15;    // row this thread selects for
    const int selseg = tid >> 4;    // 8-column segment within the 128-chunk

    // ISA A/B layout for 16x16x4 f32: VGPR0 holds K={0|2}, VGPR1 holds K={1|3};
    // lane half selects (k0,k1) = (0,1) or (2,3) -> always an adjacent pair.
    const int k0 = half ? 2 : 0;
    const float* arow = &As[n16][0];

    for (int colchunk = 0; colchunk < NTRAIN; colchunk += 128) {
        const int col = colchunk + wave * 16 + n16;  // this lane's B column
        const float* bbase = TTI + (size_t)col * 2 + (size_t)half * NTRAIN * 2;
        v8f acc = {0.f, 0.f, 0.f, 0.f, 0.f, 0.f, 0.f, 0.f};

#pragma unroll 8
        for (int kb = 0; kb < DFEAT; kb += 4) {
            v2f a = *(const v2f*)(arow + kb + k0);                       // ds_load_b64
            v2f b = *(const v2f*)(bbase + (size_t)(kb >> 1) * NTRAIN * 2); // global b64
            acc = __builtin_amdgcn_wmma_f32_16x16x4_f32(
                false, a, false, b, (short)0, acc, false, false);
        }

        // dot -> squared distance, dump to LDS
        const float tnv = tnorm[col];
#pragma unroll
        for (int r = 0; r < 8; ++r) {
            int M = r + (half ? 8 : 0);
            Dist[M][wave * 16 + n16] = qn[M] + tnv - 2.f * acc[r];
        }
        __syncthreads();

        // each thread: 8 candidates into its private sorted list
#pragma unroll
        for (int j = 0; j < 8; ++j) {
            int   c  = selseg * 8 + j;
            float d  = Dist[selrow][c];
            int   gi = colchunk + c;
            if (d < pld[tid][MAXK - 1]) {
                int pos = MAXK - 1;
                while (pos > 0 && pld[tid][pos - 1] > d) {
                    pld[tid][pos] = pld[tid][pos - 1];
                    pli[tid][pos] = pli[tid][pos - 1];
                    --pos;
                }
                pld[tid][pos] = d;
                pli[tid][pos] = gi;
            }
        }
        __syncthreads();
    }

    // merge 16 partial lists per row (threads 0..15)
    if (tid < 16) {
#pragma unroll
        for (int j = 0; j < MAXK; ++j) { topd[tid][j] = FLT_BIG; topi[tid][j] = 0; }
        for (int s = 0; s < 16; ++s) {
            int src = s * 16 + tid;   // thread with selrow==tid, selseg==s
            for (int j = 0; j < MAXK; ++j) {
                float d = pld[src][j];
                if (d >= topd[tid][MAXK - 1]) break;  // partial list is sorted
                int gi  = pli[src][j];
                int pos = MAXK - 1;
                while (pos > 0 && topd[tid][pos - 1] > d) {
                    topd[tid][pos] = topd[tid][pos - 1];
                    topi[tid][pos] = topi[tid][pos - 1];
                    --pos;
                }
                topd[tid][pos] = d;
                topi[tid][pos] = gi;
            }
        }
        for (int j = 0; j < MAXK; ++j)
            nbr[(size_t)(rowblk + tid) * MAXK + j] = topi[tid][j];
    }
}

// -------------------------------------------------------------------------
// Count tables: layout [4][NCLS][MAXK+1] ints:
//   0: c1 (k=10)   1: c0 (k=10)   2: c1 (k=20)   3: c0 (k=20)
// -------------------------------------------------------------------------
__global__ void zero_counts_kernel(int* __restrict__ counts) {
    int i = blockIdx.x * blockDim.x + threadIdx.x;
    if (i < 4 * NCLS * (MAXK + 1)) counts[i] = 0;
}

__global__ void accum_counts_kernel(const int* __restrict__ labels,
                                    const int* __restrict__ nbr_t,
                                    int* __restrict__ counts) {
    int n = blockIdx.x * blockDim.x + threadIdx.x;
    if (n >= NTRAIN) return;
    int nb[MAXK];
#pragma unroll
    for (int j = 0; j < MAXK; ++j) nb[j] = nbr_t[(size_t)n * MAXK + j];

    const int TS = NCLS * (MAXK + 1);
    for (int c = 0; c < NCLS; ++c) {
        int d10 = 0, d20 = 0;
#pragma unroll
        for (int j = 0; j < MAXK; ++j) {
            int l = labels[(size_t)nb[j] * NCLS + c];
            d20 += l;
            if (j < K_A) d10 += l;
        }
        int y = labels[(size_t)n * NCLS + c];
        atomicAdd(&counts[(y ? 0 : 1) * TS + c * (MAXK + 1) + d10], 1);
        atomicAdd(&counts[(y ? 2 : 3) * TS + c * (MAXK + 1) + d20], 1);
    }
}

// -------------------------------------------------------------------------
// Posterior: one thread per (query, class); writes out[2][NQUERY][NCLS].
// -------------------------------------------------------------------------
__global__ void posterior_kernel(const int* __restrict__ labels,
                                 const int* __restrict__ nbr_q,
                                 const int* __restrict__ counts,
                                 float* __restrict__ out) {
    int idx = blockIdx.x * blockDim.x + threadIdx.x;  // b*NCLS + c
    if (idx >= NQUERY * NCLS) return;
    int b = idx / NCLS;
    int c = idx - b * NCLS;

    int nb[MAXK];
#pragma unroll
    for (int j = 0; j < MAXK; ++j) nb[j] = nbr_q[(size_t)b * MAXK + j];
    int d10 = 0, d20 = 0;
#pragma unroll
    for (int j = 0; j < MAXK; ++j) {
        int l = labels[(size_t)nb[j] * NCLS + c];
        d20 += l;
        if (j < K_A) d10 += l;
    }

    const int TS = NCLS * (MAXK + 1);
    const int* c1_10 = counts + 0 * TS;
    const int* c0_10 = counts + 1 * TS;
    const int* c1_20 = counts + 2 * TS;
    const int* c0_20 = counts + 3 * TS;

    float cntpos = 0.f;
    for (int d = 0; d <= MAXK; ++d) cntpos += (float)c1_20[c * (MAXK + 1) + d];

    const float priorT = (1.f + cntpos) / (2.f + (float)NTRAIN);
    const float priorF = 1.f - priorT;

    {   // k = 10
        float condT = (1.f + (float)c1_10[c * (MAXK + 1) + d10]) /
                      ((float)(K_A + 1) + cntpos);
        float condF = (1.f + (float)c0_10[c * (MAXK + 1) + d10]) /
                      ((float)(K_A + 1) + ((float)NTRAIN - cntpos));
        float pt = priorT * condT, pf = priorF * condF;
        out[0 * NQUERY * NCLS + idx] = pt / (pt + pf);
    }
    {   // k = 20
        float condT = (1.f + (float)c1_20[c * (MAXK + 1) + d20]) /
                      ((float)(K_B + 1) + cntpos);
        float condF = (1.f + (float)c0_20[c * (MAXK + 1) + d20]) /
                      ((float)(K_B + 1) + ((float)NTRAIN - cntpos));
        float pt = priorT * condT, pf = priorF * condF;
        out[1 * NQUERY * NCLS + idx] = pt / (pt + pf);
    }
}

// -------------------------------------------------------------------------
extern "C" void kernel_launch(void* const* d_in, const int* in_sizes, int n_in,
                              void* d_out, int out_size, void* d_ws, size_t ws_size,
                              hipStream_t stream) {
    (void)in_sizes; (void)n_in; (void)out_size; (void)ws_size;
    const float* feats  = (const float*)d_in[0];   // [NQUERY, DFEAT]
    const float* trainf = (const float*)d_in[1];   // [NTRAIN, DFEAT]
    const int*   labels = (const int*)d_in[2];     // [NTRAIN, NCLS]

    // workspace layout (256B-aligned chunks)
    char* ws = (char*)d_ws;
    size_t off = 0;
    auto carve = [&](size_t bytes) {
        void* p = ws + off;
        off += (bytes + 255) & ~(size_t)255;
        return p;
    };
    float* TTI    = (float*)carve((size_t)DFEAT * NTRAIN * sizeof(float)); // 8 MB
    float* tnorm  = (float*)carve((size_t)NTRAIN * sizeof(float));
    float* qnorm  = (float*)carve((size_t)NQUERY * sizeof(float));
    int*   nbr_t  = (int*)  carve((size_t)NTRAIN * MAXK * sizeof(int));
    int*   nbr_q  = (int*)  carve((size_t)NQUERY * MAXK * sizeof(int));
    int*   counts = (int*)  carve((size_t)4 * NCLS * (MAXK + 1) * sizeof(int));

    // 1) pair-interleaved transpose + norms
    interleave_kernel<<<((DFEAT / 2) * NTRAIN + 255) / 256, 256, 0, stream>>>(trainf, TTI);
    row_norms_kernel<<<(NTRAIN + 255) / 256, 256, 0, stream>>>(trainf, tnorm, NTRAIN);
    row_norms_kernel<<<(NQUERY + 255) / 256, 256, 0, stream>>>(feats, qnorm, NQUERY);

    // 2) fused WMMA distances + top-20 (train-vs-train, then query-vs-train)
    knn_topk_kernel<<<NTRAIN / 16, 256, 0, stream>>>(trainf, tnorm, TTI, tnorm, nbr_t);
    knn_topk_kernel<<<NQUERY / 16, 256, 0, stream>>>(feats, qnorm, TTI, tnorm, nbr_q);

    // 3) MLkNN statistics + posterior
    zero_counts_kernel<<<(4 * NCLS * (MAXK + 1) + 255) / 256, 256, 0, stream>>>(counts);
    accum_counts_kernel<<<(NTRAIN + 255) / 256, 256, 0, stream>>>(labels, nbr_t, counts);
    posterior_kernel<<<(NQUERY * NCLS + 255) / 256, 256, 0, stream>>>(labels, nbr_q,
                                                                      counts, (float*)d_out);
}